// AttentionConv_63617055589076
// MI455X (gfx1250) — compile-verified
//
#include <hip/hip_runtime.h>

typedef __attribute__((ext_vector_type(2))) float v2f;
typedef __attribute__((ext_vector_type(8))) float v8f;

#define HWp 4096      // H*W
#define Hd  64
#define Wdm 64
#define Cc  128       // input channels
#define Oc  128       // output channels
#define KK  7
#define PADk 3
#define Mtot 16384    // B*H*W

// ---------------------------------------------------------------------------
// Kernel 1: 1x1-conv projections as GEMM via V_WMMA_F32_16X16X4_F32.
//   Y[m,n] = sum_k X[b,k,p] * W[n,k],  m=(b,p), M=16384, N=128, K=128
//   Output layout NHWC: Y[m*128 + n]  (q/k/v staging buffers in d_ws)
// blockIdx.x : M tile (16 rows). wave (of 8) : N tile. blockIdx.y : {q,k,v}.
// ---------------------------------------------------------------------------
__global__ __launch_bounds__(256) void proj_gemm_wmma(
    const float* __restrict__ X,
    const float* __restrict__ Wq, const float* __restrict__ Wk,
    const float* __restrict__ Wv,
    float* __restrict__ Yq, float* __restrict__ Yk, float* __restrict__ Yv)
{
    const int wave = threadIdx.x >> 5;      // 0..7 -> N tile
    const int lane = threadIdx.x & 31;
    const int half = lane >> 4;             // 0: lanes 0-15, 1: lanes 16-31
    const int r    = lane & 15;

    const int m0 = blockIdx.x * 16;
    const int n0 = wave * 16;
    const int b  = m0 / HWp;                // tiles never straddle a batch
    const int p  = m0 - b * HWp;

    const float* Wsel; float* Ysel;
    if (blockIdx.y == 0)      { Wsel = Wq; Ysel = Yq; }
    else if (blockIdx.y == 1) { Wsel = Wk; Ysel = Yk; }
    else                      { Wsel = Wv; Ysel = Yv; }

    // A element [m0+r, k] lives at X[b*C*HW + k*HW + (p+r)]
    const float* Abase = X + (size_t)b * Cc * HWp + p + r;
    // B element [k, n0+r] lives at W[(n0+r)*C + k]
    const float* Bbase = Wsel + (size_t)(n0 + r) * Cc;

    v8f acc = {};
    #pragma unroll
    for (int ks = 0; ks < Cc; ks += 4) {
        // fp32 16x4 A layout: VGPR0 holds K=k0 (lanes 0-15) / K=k0+2 (16-31),
        // VGPR1 holds K=k0+1 / K=k0+3. B layout is the symmetric transpose.
        const int k0 = ks + 2 * half;
        v2f a, bf;
        a.x  = Abase[(size_t)k0 * HWp];
        a.y  = Abase[(size_t)(k0 + 1) * HWp];
        bf.x = Bbase[k0];
        bf.y = Bbase[k0 + 1];
        acc = __builtin_amdgcn_wmma_f32_16x16x4_f32(
                  false, a, false, bf, (short)0, acc, false, false);
    }

    // C/D 16x16 f32 layout: VGPR rr -> row (rr + 8*half), col = lane&15.
    float* Yp = Ysel + (size_t)(m0 + 8 * half) * Oc + n0 + r;
    #pragma unroll
    for (int rr = 0; rr < 8; ++rr)
        Yp[(size_t)rr * Oc] = acc[rr];
}

// ---------------------------------------------------------------------------
// Kernel 2: local windowed attention. One thread = one (b,h,w,head).
// Heads packed into low 3 bits of tid so a wave covers 4 pixels x 8 heads
// => contiguous 128-channel rows of NHWC q/k/v (coalesced).
// Zero-padded k/v handled implicitly: OOB taps score = q.rel only, no V.
// ---------------------------------------------------------------------------
__global__ __launch_bounds__(256) void local_attn(
    const float* __restrict__ Q, const float* __restrict__ Kb,
    const float* __restrict__ Vb,
    const float* __restrict__ rel_h, const float* __restrict__ rel_w,
    float* __restrict__ out)
{
    const int t  = blockIdx.x * blockDim.x + threadIdx.x;   // [0, B*HW*8)
    const int g  = t & 7;
    const int pp = t >> 3;              // pixel index in [0, B*HW)
    const int b  = pp >> 12;            // / 4096
    const int hw = pp & 4095;
    const int h  = hw >> 6;
    const int w  = hw & 63;

    // q vector for this head (16 contiguous floats)
    const float* qp = Q + (size_t)pp * Oc + g * 16;
    float q[16];
    #pragma unroll
    for (int c = 0; c < 16; ++c) q[c] = qp[c];

    // Precompute q . rel  (head fully inside rel_h half iff g < 4)
    const bool useH = (g < 4);
    const float* rel = useH ? (rel_h + (size_t)g * 16 * KK)
                            : (rel_w + (size_t)(g - 4) * 16 * KK);
    float qr[KK];
    #pragma unroll
    for (int o = 0; o < KK; ++o) {
        float s = 0.f;
        #pragma unroll
        for (int c = 0; c < 16; ++c) s += q[c] * rel[c * KK + o];
        qr[o] = s;
    }

    // Pass 1: scores over the 7x7 window
    float sc[KK * KK];
    float smax = -3.402823466e38f;
    #pragma unroll
    for (int i = 0; i < KK; ++i) {
        const int hh = h + i - PADk;
        #pragma unroll
        for (int j = 0; j < KK; ++j) {
            const int ww = w + j - PADk;
            float s = useH ? qr[i] : qr[j];
            if (hh >= 0 && hh < Hd && ww >= 0 && ww < Wdm) {
                const float* kp = Kb + ((size_t)(b * HWp + hh * Wdm + ww)) * Oc + g * 16;
                #pragma unroll
                for (int c = 0; c < 16; ++c) s += q[c] * kp[c];
            }
            sc[i * KK + j] = s;
            smax = fmaxf(smax, s);
        }
    }

    float ssum = 0.f;
    #pragma unroll
    for (int n = 0; n < KK * KK; ++n) { sc[n] = __expf(sc[n] - smax); ssum += sc[n]; }
    const float inv = 1.0f / ssum;

    // Pass 2: probability-weighted V accumulation
    float acc[16];
    #pragma unroll
    for (int c = 0; c < 16; ++c) acc[c] = 0.f;
    #pragma unroll
    for (int i = 0; i < KK; ++i) {
        const int hh = h + i - PADk;
        #pragma unroll
        for (int j = 0; j < KK; ++j) {
            const int ww = w + j - PADk;
            if (hh >= 0 && hh < Hd && ww >= 0 && ww < Wdm) {
                const float pw = sc[i * KK + j] * inv;
                const float* vp = Vb + ((size_t)(b * HWp + hh * Wdm + ww)) * Oc + g * 16;
                #pragma unroll
                for (int c = 0; c < 16; ++c) acc[c] += pw * vp[c];
            }
        }
    }

    // out[b, g*16+c, h, w]   (reference output layout [B,O,H,W])
    float* op = out + ((size_t)b * Oc + g * 16) * HWp + hw;
    #pragma unroll
    for (int c = 0; c < 16; ++c) op[(size_t)c * HWp] = acc[c];
}

// ---------------------------------------------------------------------------
extern "C" void kernel_launch(void* const* d_in, const int* in_sizes, int n_in,
                              void* d_out, int out_size, void* d_ws, size_t ws_size,
                              hipStream_t stream) {
    (void)in_sizes; (void)n_in; (void)out_size; (void)ws_size;
    const float* x     = (const float*)d_in[0];
    const float* Wq    = (const float*)d_in[1];
    const float* Wk    = (const float*)d_in[2];
    const float* Wv    = (const float*)d_in[3];
    const float* rel_h = (const float*)d_in[4];
    const float* rel_w = (const float*)d_in[5];
    float* out = (float*)d_out;

    // workspace: q | k | v, each M*O floats = 8 MB (24 MB total)
    float* q = (float*)d_ws;
    float* k = q + (size_t)Mtot * Oc;
    float* v = k + (size_t)Mtot * Oc;

    dim3 gemm_grid(Mtot / 16, 3);
    proj_gemm_wmma<<<gemm_grid, 256, 0, stream>>>(x, Wq, Wk, Wv, q, k, v);

    const int threads = 4 * HWp * 8;   // B*HW*heads = 131072
    local_attn<<<threads / 256, 256, 0, stream>>>(q, k, v, rel_h, rel_w, out);
}